// CoA_SSI_DDI_31963146616866
// MI455X (gfx1250) — compile-verified
//
#include <hip/hip_runtime.h>

// Problem constants (fixed by the reference: B=8, Nc=Nd=1024, D=64, H=32)
#define BATCH 8
#define NC 1024
#define ND 1024
#define DD 64
#define HH 32
#define ROWS (BATCH * NC)   // 8192 rows for each projection

typedef float v2f __attribute__((ext_vector_type(2)));
typedef float v8f __attribute__((ext_vector_type(8)));

// ---------------------------------------------------------------------------
// Projection: Y[r, 0:32] = X[r, 0:64] @ W[0:64, 0:32] (+ bias), via
// V_WMMA_F32_16X16X4_F32. One wave computes a 16x32 tile: 2 N-tiles of 16,
// 16 K-steps of 4. Operand layouts per CDNA5 ISA 7.12.2 (32-bit matrices):
//   A (16x4):  lane l: v0=A[l&15][k0+2*(l>>4)], v1=A[l&15][k0+2*(l>>4)+1]
//   B (4x16):  lane l: v0=B[k0+2*(l>>4)][l&15], v1=B[k0+2*(l>>4)+1][l&15]
//   C/D (16x16): reg r, lane l -> row r+8*(l>>4), col l&15
// ---------------------------------------------------------------------------
__global__ __launch_bounds__(128)
void proj_wmma_f32(const float* __restrict__ X,   // [ROWS, 64]
                   const float* __restrict__ W,   // [64, 32]
                   const float* __restrict__ bias,// [32]
                   float* __restrict__ Y,         // [ROWS, 32]
                   int addBias) {
  const int lane = threadIdx.x & 31;
  const int wave = threadIdx.x >> 5;
  const int row0 = (blockIdx.x * 4 + wave) * 16;
  const int m    = lane & 15;
  const int kk   = (lane >> 4) << 1;          // 0 or 2

  v8f acc0 = {};
  v8f acc1 = {};
  const float* __restrict__ xrow = X + (size_t)(row0 + m) * DD;

#pragma unroll
  for (int k0 = 0; k0 < DD; k0 += 4) {
    const int k = k0 + kk;
    v2f A;
    A.x = xrow[k];
    A.y = xrow[k + 1];                         // contiguous pair -> b64 load
    v2f B0, B1;
    B0.x = W[(size_t)k * HH + m];
    B0.y = W[(size_t)(k + 1) * HH + m];
    B1.x = W[(size_t)k * HH + 16 + m];
    B1.y = W[(size_t)(k + 1) * HH + 16 + m];
    // 8 args: (neg_a, A, neg_b, B, c_mod, C, reuse_a, reuse_b)
    acc0 = __builtin_amdgcn_wmma_f32_16x16x4_f32(false, A, false, B0,
                                                 (short)0, acc0, false, false);
    acc1 = __builtin_amdgcn_wmma_f32_16x16x4_f32(false, A, false, B1,
                                                 (short)0, acc1, false, false);
  }

  const float b0 = addBias ? bias[m] : 0.0f;
  const float b1 = addBias ? bias[16 + m] : 0.0f;
#pragma unroll
  for (int r = 0; r < 8; ++r) {
    const int row = row0 + r + ((lane >> 4) << 3);
    Y[(size_t)row * HH + m]      = acc0[r] + b0;
    Y[(size_t)row * HH + 16 + m] = acc1[r] + b1;
  }
}

// ---------------------------------------------------------------------------
// Fused broadcast + tanh + reduce:
//   out[b,i,j] = sum_h a[h] * tanh(cA[b,i,h] + dA[b,j,h])   (bias in cA)
// Block = 256 threads, tile = 16 i x 256 j. cA tile in LDS (broadcast reads),
// a[] hoisted to registers, dA row per-lane in registers.
// tanh(x) = 1 - 2*rcp(exp2(2*log2e*x) + 1)  -> v_exp_f32 + v_rcp_f32,
// saturates correctly for large |x| (exp2->inf -> rcp->0 -> 1; exp2->0 -> -1).
// ---------------------------------------------------------------------------
__global__ __launch_bounds__(256)
void coattn_main(const float* __restrict__ cA,   // [B*NC, 32], bias folded in
                 const float* __restrict__ dA,   // [B*ND, 32]
                 const float* __restrict__ a,    // [32]
                 float* __restrict__ out) {      // [B, NC, ND]
  __shared__ float s_c[16 * HH];                 // 2 KB cell tile

  const int tid = threadIdx.x;
  const int b  = blockIdx.z;
  const int i0 = blockIdx.y * 16;
  const int j  = blockIdx.x * 256 + tid;

  // Stage the 16x32 cell tile into LDS (512 floats / 256 threads).
  {
    const float* __restrict__ src = cA + ((size_t)b * NC + i0) * HH;
    s_c[tid]       = src[tid];
    s_c[tid + 256] = src[tid + 256];
  }
  __syncthreads();

  // Hoist a[] (uniform) and this lane's drug row into registers.
  float areg[HH];
#pragma unroll
  for (int h = 0; h < HH; ++h) areg[h] = a[h];   // uniform -> s_loads

  float dreg[HH];
  {
    const float4* __restrict__ dsrc =
        (const float4*)(dA + ((size_t)b * ND + j) * HH);
#pragma unroll
    for (int q = 0; q < 8; ++q) {
      const float4 v = dsrc[q];
      dreg[4 * q + 0] = v.x;
      dreg[4 * q + 1] = v.y;
      dreg[4 * q + 2] = v.z;
      dreg[4 * q + 3] = v.w;
    }
  }

  float* __restrict__ orow = out + (((size_t)b * NC + i0) * ND + j);
  const float TWO_LOG2E = 2.8853900817779268f;   // 2 / ln(2)

  for (int i = 0; i < 16; ++i) {                 // rolled: keep I$ small
    float acc = 0.0f;
#pragma unroll
    for (int h = 0; h < HH; ++h) {
      const float x = s_c[i * HH + h] + dreg[h];
      const float t = __builtin_amdgcn_exp2f(x * TWO_LOG2E);
      const float r = __builtin_amdgcn_rcpf(t + 1.0f);
      acc = fmaf(areg[h], fmaf(-2.0f, r, 1.0f), acc);
    }
    orow[(size_t)i * ND] = acc;
  }
}

// ---------------------------------------------------------------------------
// Launch: two WMMA projections into workspace, then the fused kernel.
// d_in order: cell, drug, w_q, w_k, bias, a. NOTE: cell uses w_k, drug uses w_q.
// Workspace: cA (1 MB) + dA (1 MB).
// ---------------------------------------------------------------------------
extern "C" void kernel_launch(void* const* d_in, const int* in_sizes, int n_in,
                              void* d_out, int out_size, void* d_ws, size_t ws_size,
                              hipStream_t stream) {
  const float* cell = (const float*)d_in[0];
  const float* drug = (const float*)d_in[1];
  const float* w_q  = (const float*)d_in[2];
  const float* w_k  = (const float*)d_in[3];
  const float* bias = (const float*)d_in[4];
  const float* a    = (const float*)d_in[5];
  float* out = (float*)d_out;

  float* cA = (float*)d_ws;                       // [ROWS, 32]
  float* dA = cA + (size_t)ROWS * HH;             // [ROWS, 32]

  // 128 blocks x 4 waves x 16 rows = 8192 rows each.
  proj_wmma_f32<<<ROWS / 64, 128, 0, stream>>>(cell, w_k, bias, cA, 1);
  proj_wmma_f32<<<ROWS / 64, 128, 0, stream>>>(drug, w_q, bias, dA, 0);

  dim3 grid(ND / 256, NC / 16, BATCH);
  coattn_main<<<grid, 256, 0, stream>>>(cA, dA, a, out);
}